// Reproj_30399778521134
// MI455X (gfx1250) — compile-verified
//
#include <hip/hip_runtime.h>
#include <stdint.h>
#include <math.h>

// ---------------------------------------------------------------------------
// Bundle-adjustment reprojection residuals, MI455X (gfx1250).
//
// Memory-bound (AI < 1 flop/byte): 192 MB streaming + L2-resident gather
// tables => ~8-10us at 23.3 TB/s. No matmul structure => WMMA inapplicable.
// CDNA5 features used instead:
//   * TDM tensor_load_to_lds: DMA the 80KB camera table into LDS per block
//     (320KB LDS/WGP -> 4 blocks/WGP still fit), waited via s_wait_tensorcnt.
//   * Quaternions normalized ONCE per camera in LDS (each cam reused ~4000x).
//   * b128 coalesced streams (4 obs/thread), global_prefetch on index streams.
// ---------------------------------------------------------------------------

typedef unsigned int v4u __attribute__((ext_vector_type(4)));
typedef int          v8i __attribute__((ext_vector_type(8)));
typedef int          v4i __attribute__((ext_vector_type(4)));

union TdmG0 { uint64_t u64[2]; v4u v; };
union TdmG1 { uint64_t u64[4]; v8i v; };
union TdmG2 { uint64_t u64[2]; v4i v; };

#define BLOCK_THREADS 256
#define OBS_PER_THREAD 4

// Project one observation. `s_cam` points into LDS (10 floats/cam, quat pre-normalized).
__device__ __forceinline__ void project_one(const float* __restrict__ s_cam,
                                            const float* __restrict__ points_3d,
                                            int ci, int pi, float p2x, float p2y,
                                            float& ox, float& oy)
{
    const float* c = s_cam + ci * 10;          // ds_load from LDS
    const float w  = c[0], vx = c[1], vy = c[2], vz = c[3];
    const float* P = points_3d + 3 * pi;       // L2-resident gather
    const float px = P[0], py = P[1], pz = P[2];

    // t = cross(v, p) + w*p
    const float tx = fmaf(vy, pz, fmaf(-vz, py, w * px));
    const float ty = fmaf(vz, px, fmaf(-vx, pz, w * py));
    const float tz = fmaf(vx, py, fmaf(-vy, px, w * pz));
    // rotated = p + 2*cross(v, t), then translate
    const float rx = fmaf(2.0f, fmaf(vy, tz, -vz * ty), px) + c[4];
    const float ry = fmaf(2.0f, fmaf(vz, tx, -vx * tz), py) + c[5];
    const float rz = fmaf(2.0f, fmaf(vx, ty, -vy * tx), pz) + c[6];

    // proj = -p.xy / p.z
    const float inv = -1.0f / rz;
    const float qx = rx * inv, qy = ry * inv;

    const float n  = fmaf(qx, qx, qy * qy);
    const float f = c[7], k1 = c[8], k2 = c[9];
    const float s  = fmaf(fmaf(k2, n * n, k1 * n), f, f);  // (1 + k1 n + k2 n^2) * f
    ox = fmaf(qx, s, -p2x);
    oy = fmaf(qy, s, -p2y);
}

__global__ __launch_bounds__(BLOCK_THREADS)
void Reproj_30399778521134_kernel(const float* __restrict__ points_2d,
                                  const int*   __restrict__ camera_indices,
                                  const int*   __restrict__ point_indices,
                                  const float* __restrict__ camera_params,
                                  const float* __restrict__ points_3d,
                                  float*       __restrict__ out,
                                  int n_obs, int n_cam)
{
    extern __shared__ float s_cam[];                 // n_cam*10 floats (80 KB)
    const int cam_floats = n_cam * 10;

#if __has_builtin(__builtin_amdgcn_tensor_load_to_lds) && __has_builtin(__builtin_amdgcn_s_wait_tensorcnt)
    if (threadIdx.x < 32) {
        // --- Build Tensor DMA Descriptor (D#): 1-row 2D tile of cam_floats dwords ---
        const uint64_t ga   = (uint64_t)(uintptr_t)camera_params;
        const uint64_t dim0 = (uint64_t)(uint32_t)cam_floats;    // 20000 dwords

        TdmG0 g0;
        g0.u64[0] = 1ull;                                        // count=1, lds_addr=0 (sole LDS alloc)
        g0.u64[1] = (ga & 0x01FFFFFFFFFFFFFFull) | (2ull << 62); // global_addr[56:0], type=2
        TdmG1 g1;
        // [17:16]=data_size(4B)=2 ; [63:48]=tensor_dim0[15:0]
        g1.u64[0] = 0x20000ull | ((dim0 & 0xFFFFull) << 48);
        // [15:0]=tensor_dim0[31:16] ; [47:16]=tensor_dim1=1 ; [63:48]=tile_dim0
        g1.u64[1] = ((dim0 >> 16) & 0xFFFFull) | (1ull << 16) | ((dim0 & 0xFFFFull) << 48);
        // [15:0]=tile_dim1=1 ; [31:16]=tile_dim2=0 ; [63:32]=tensor_dim0_stride
        g1.u64[2] = 1ull | (dim0 << 32);
        // [15:0]=tensor_dim0_stride[47:32]=0 ; [63:16]=tensor_dim1_stride
        g1.u64[3] = (dim0 << 16);
        TdmG2 g2; g2.u64[0] = 0; g2.u64[1] = 0;                  // D# group2 (unused dims, 2D tile)
        TdmG2 g3; g3.u64[0] = 0; g3.u64[1] = 0;                  // D# group3 (unused)
        TdmG1 g4; g4.u64[0] = 0; g4.u64[1] = 0; g4.u64[2] = 0; g4.u64[3] = 0; // trailing operand (clang-23 6-arg form)

        // amdgpu-toolchain (clang-23) signature:
        // (uint32x4 g0, int32x8 g1, int32x4, int32x4, int32x8, i32 cpol)
        __builtin_amdgcn_tensor_load_to_lds(g0.v, g1.v, g2.v, g3.v, g4.v, 0);
        __builtin_amdgcn_s_wait_tensorcnt(0);
    }
#else
    for (int i = threadIdx.x; i < cam_floats; i += BLOCK_THREADS)
        s_cam[i] = camera_params[i];
#endif
    __syncthreads();

    // Normalize each camera quaternion once per block (amortized ~4000x reuse).
    for (int c = threadIdx.x; c < n_cam; c += BLOCK_THREADS) {
        float* q = s_cam + c * 10;
        const float q0 = q[0], q1 = q[1], q2 = q[2], q3 = q[3];
        const float inv = 1.0f / sqrtf(fmaf(q0, q0, fmaf(q1, q1, fmaf(q2, q2, q3 * q3))));
        q[0] = q0 * inv; q[1] = q1 * inv; q[2] = q2 * inv; q[3] = q3 * inv;
    }
    __syncthreads();

    const int stride = gridDim.x * BLOCK_THREADS * OBS_PER_THREAD;
    for (int base = (blockIdx.x * BLOCK_THREADS + threadIdx.x) * OBS_PER_THREAD;
         base < n_obs; base += stride)
    {
        // Prefetch next stride of the streaming index arrays (global_prefetch_b8;
        // speculative TH: OOB translations silently dropped).
        __builtin_prefetch(camera_indices + base + stride, 0, 1);
        __builtin_prefetch(point_indices  + base + stride, 0, 1);

        if (base + OBS_PER_THREAD <= n_obs) {
            const int4   ci = *(const int4*)(camera_indices + base);   // b128
            const int4   pi = *(const int4*)(point_indices + base);    // b128
            const float4 pa = *(const float4*)(points_2d + 2 * base);      // obs 0,1
            const float4 pb = *(const float4*)(points_2d + 2 * base + 4);  // obs 2,3

            float4 oa, ob;
            project_one(s_cam, points_3d, ci.x, pi.x, pa.x, pa.y, oa.x, oa.y);
            project_one(s_cam, points_3d, ci.y, pi.y, pa.z, pa.w, oa.z, oa.w);
            project_one(s_cam, points_3d, ci.z, pi.z, pb.x, pb.y, ob.x, ob.y);
            project_one(s_cam, points_3d, ci.w, pi.w, pb.z, pb.w, ob.z, ob.w);

            *(float4*)(out + 2 * base)     = oa;   // b128 coalesced stores
            *(float4*)(out + 2 * base + 4) = ob;
        } else {
            for (int o = base; o < n_obs; ++o) {
                float ox, oy;
                project_one(s_cam, points_3d, camera_indices[o], point_indices[o],
                            points_2d[2 * o], points_2d[2 * o + 1], ox, oy);
                out[2 * o]     = ox;
                out[2 * o + 1] = oy;
            }
        }
    }
}

extern "C" void kernel_launch(void* const* d_in, const int* in_sizes, int n_in,
                              void* d_out, int out_size, void* d_ws, size_t ws_size,
                              hipStream_t stream) {
    const float* points_2d      = (const float*)d_in[0];
    const int*   camera_indices = (const int*)d_in[1];
    const int*   point_indices  = (const int*)d_in[2];
    const float* camera_params  = (const float*)d_in[3];
    const float* points_3d      = (const float*)d_in[4];
    float*       out            = (float*)d_out;

    const int n_obs = in_sizes[1];          // camera_indices element count
    const int n_cam = in_sizes[3] / 10;     // camera_params rows

    const int per_block = BLOCK_THREADS * OBS_PER_THREAD;
    int blocks = (n_obs + per_block - 1) / per_block;
    if (blocks > 2048) blocks = 2048;       // grid-stride; amortizes 80KB TDM staging
    if (blocks < 1) blocks = 1;

    const size_t lds_bytes = (size_t)n_cam * 10 * sizeof(float);  // 80 KB dynamic LDS

    Reproj_30399778521134_kernel<<<blocks, BLOCK_THREADS, lds_bytes, stream>>>(
        points_2d, camera_indices, point_indices, camera_params, points_3d,
        out, n_obs, n_cam);
}